// TransportMapKernel_6116033429507
// MI455X (gfx1250) — compile-verified
//
#include <hip/hip_runtime.h>
#include <hip/hip_bf16.h>
#include <math.h>

typedef __attribute__((ext_vector_type(2))) float v2f;
typedef __attribute__((ext_vector_type(8))) float v8f;

#define MDIM 30
#define XS_STRIDE 33   // 64 x 32 (K padded), stride 33 to avoid bank conflicts
#define G_STRIDE 65    // 64 x 64, stride 65

__global__ __launch_bounds__(256) void tmk_kernel(
    const float* __restrict__ aug,          // (64, N, 31)
    const float* __restrict__ scales,       // (N)
    const float* __restrict__ nug_mean,     // (N)
    const float* __restrict__ theta_q,      // (1)
    const float* __restrict__ sigma_params, // (2)
    const float* __restrict__ lengthscale,  // (1)
    const long long* __restrict__ batch_idx,// (N) int64
    float* __restrict__ out,                // (2, N, 64, 64): [g ; chol]
    int N)
{
    __shared__ float xs[64 * XS_STRIDE];
    __shared__ float g[64 * G_STRIDE];

    const int tid = threadIdx.x;
    const int b   = blockIdx.x;

    // ---- async-stage raw aug rows into LDS (GLOBAL_LOAD_ASYNC_TO_LDS_B32, ASYNCcnt) ----
    {
        const int c  = tid & 31;        // 0..31
        const int r0 = tid >> 5;        // 0..7
        #pragma unroll
        for (int pass = 0; pass < 8; ++pass) {
            const int r = pass * 8 + r0;
            if (c < MDIM) {
                const float* gp = &aug[((size_t)r * (size_t)N + (size_t)b) * 31 + 1 + c];
                // per-lane LDS byte address = low 32 bits of flat shared pointer
                unsigned lds_off = (unsigned)(unsigned long long)(const void*)&xs[r * XS_STRIDE + c];
                asm volatile("global_load_async_to_lds_b32 %0, %1, off"
                             :: "v"(lds_off), "v"(gp) : "memory");
            } else {
                xs[r * XS_STRIDE + c] = 0.0f;   // zero K-pad columns 30,31
            }
        }
        asm volatile("s_wait_asynccnt 0" ::: "memory");
    }

    // ---- scalar prep (uniform) — runs after the async wait above ----
    const float et      = __expf(theta_q[0]);
    const float sp0     = sigma_params[0];
    const float sp1     = sigma_params[1];
    const float inv_ls2 = 1.0f / (3.0f * __expf(2.0f * lengthscale[0]));  // ls^2 = 3*exp(2*l)
    const float sc      = scales[b];
    const float sigma   = __expf(sp0 + sp1 * __logf(sc));
    const float sig2    = sigma * sigma;
    const float inv_nug = 1.0f / nug_mean[b];
    const bool  is0     = (batch_idx[b] == 0);

    __syncthreads();

    // ---- in-place transform: NaN -> 0, apply column scaling exp(-0.5*(c+1)*exp(theta)) ----
    {
        const int   c    = tid & 31;
        const int   r0   = tid >> 5;
        const float scal = __expf(-0.5f * (float)(c + 1) * et);
        if (c < MDIM) {
            #pragma unroll
            for (int pass = 0; pass < 8; ++pass) {
                const int r = pass * 8 + r0;
                float t = xs[r * XS_STRIDE + c];
                xs[r * XS_STRIDE + c] = (t != t) ? 0.0f : t * scal;
            }
        }
    }
    __syncthreads();

    // ---- gram: linear = xs * xs^T via f32 WMMA, 16 tiles of 16x16, 2 tiles per wave ----
    const int lane = tid & 31;
    const int wave = tid >> 5;
    const int lr   = lane & 15;
    const int hi   = lane >> 4;   // 0: K={0,1}; 1: K={2,3} within each K4 step
    v8f acc[2];
    #pragma unroll
    for (int p = 0; p < 2; ++p) {
        const int t  = wave * 2 + p;
        const int tr = t >> 2;
        const int tc = t & 3;
        v8f c = {};
        #pragma unroll
        for (int kk = 0; kk < 8; ++kk) {       // K = 32 in steps of 4
            const int kb = kk * 4 + 2 * hi;
            v2f a, bb;
            a.x  = xs[(tr * 16 + lr) * XS_STRIDE + kb];
            a.y  = xs[(tr * 16 + lr) * XS_STRIDE + kb + 1];
            bb.x = xs[(tc * 16 + lr) * XS_STRIDE + kb];   // B = xs^T: col tile rows
            bb.y = xs[(tc * 16 + lr) * XS_STRIDE + kb + 1];
            c = __builtin_amdgcn_wmma_f32_16x16x4_f32(
                    /*neg_a=*/false, a, /*neg_b=*/false, bb,
                    /*c_mod=*/(short)0, c, /*reuse_a=*/false, /*reuse_b=*/false);
        }
        acc[p] = c;
    }
    // spill accumulators (raw linear) into LDS g
    #pragma unroll
    for (int p = 0; p < 2; ++p) {
        const int t  = wave * 2 + p;
        const int tr = t >> 2;
        const int tc = t & 3;
        #pragma unroll
        for (int v = 0; v < 8; ++v)
            g[(tr * 16 + v + 8 * hi) * G_STRIDE + tc * 16 + lr] = acc[p][v];
    }
    __syncthreads();

    // ---- fused epilogue: g = (linear + sig2*matern15(d)) / nug + I  (or I for batch 0) ----
    const int gi  = tid >> 2;          // row, each thread owns 16 contiguous cols
    const int gj0 = (tid & 3) * 16;
    float gv[16];
    {
        const float s3  = 1.7320508075688772f;
        const float sqi = g[gi * G_STRIDE + gi] * inv_ls2;
        #pragma unroll
        for (int q = 0; q < 16; ++q) {
            const int   j   = gj0 + q;
            const float lv  = g[gi * G_STRIDE + j];
            const float sqj = g[j * G_STRIDE + j] * inv_ls2;
            const float d2  = sqi + sqj - 2.0f * lv * inv_ls2;
            const float d   = sqrtf(fmaxf(d2, 0.0f) + 1e-12f);
            const float nl  = (1.0f + s3 * d) * __expf(-s3 * d);
            const float eye = (gi == j) ? 1.0f : 0.0f;
            gv[q] = is0 ? eye : ((lv + sig2 * nl) * inv_nug + eye);
        }
    }
    __syncthreads();   // all reads of raw linear complete before overwrite

    // write g to LDS (for cholesky) and to global (b128 stores)
    {
        #pragma unroll
        for (int q = 0; q < 16; ++q)
            g[gi * G_STRIDE + gj0 + q] = gv[q];
        float4* go = (float4*)(out + (size_t)b * 4096 + (size_t)gi * 64 + gj0);
        #pragma unroll
        for (int q4 = 0; q4 < 4; ++q4) {
            float4 f4 = make_float4(gv[q4*4+0], gv[q4*4+1], gv[q4*4+2], gv[q4*4+3]);
            go[q4] = f4;
        }
    }
    __syncthreads();

    // ---- in-LDS 64x64 Cholesky (lower), 256 threads: row = tid&63, 4-way col split ----
    const int cx = tid & 63;
    const int cy = tid >> 6;
    for (int k = 0; k < 64; ++k) {
        if (tid == 0)
            g[k * G_STRIDE + k] = sqrtf(g[k * G_STRIDE + k]);
        __syncthreads();
        if (cy == 0 && cx > k)
            g[cx * G_STRIDE + k] /= g[k * G_STRIDE + k];
        __syncthreads();
        if (cx > k) {
            const float lik = g[cx * G_STRIDE + k];
            for (int j = k + 1 + cy; j <= cx; j += 4)
                g[cx * G_STRIDE + j] -= lik * g[j * G_STRIDE + k];
        }
        __syncthreads();
    }

    // ---- write chol (zero upper triangle) ----
    {
        float* co = out + (size_t)N * 4096 + (size_t)b * 4096 + (size_t)gi * 64 + gj0;
        #pragma unroll
        for (int q4 = 0; q4 < 4; ++q4) {
            float f[4];
            #pragma unroll
            for (int e = 0; e < 4; ++e) {
                const int j = gj0 + q4 * 4 + e;
                f[e] = (j <= gi) ? g[gi * G_STRIDE + j] : 0.0f;
            }
            *(float4*)(co + q4 * 4) = make_float4(f[0], f[1], f[2], f[3]);
        }
    }
}

extern "C" void kernel_launch(void* const* d_in, const int* in_sizes, int n_in,
                              void* d_out, int out_size, void* d_ws, size_t ws_size,
                              hipStream_t stream) {
    const float*     aug    = (const float*)d_in[0];
    const float*     scales = (const float*)d_in[1];
    const float*     nug    = (const float*)d_in[2];
    const float*     theta  = (const float*)d_in[3];
    const float*     sigp   = (const float*)d_in[4];
    const float*     lsc    = (const float*)d_in[5];
    const long long* bidx   = (const long long*)d_in[6];
    float*           out    = (float*)d_out;

    const int N = in_sizes[1];   // scales has N elements (N = 16384)
    tmk_kernel<<<dim3(N), dim3(256), 0, stream>>>(aug, scales, nug, theta, sigp, lsc, bidx, out, N);
}